// HDLoss_42726334661147
// MI455X (gfx1250) — compile-verified
//
#include <hip/hip_runtime.h>
#include <hip/hip_bf16.h>

// ---------------- problem constants (from reference setup_inputs) ------------
constexpr int Bn = 8, Cc = 4, Hh = 384, Ww = 384;
constexpr int SL = Bn * Cc;                        // 32 slices
constexpr int ROWS = SL * Hh;                      // 12288 rows
constexpr long long NTOT = (long long)SL * Hh * Ww; // 4718592 (divisible by 64)
constexpr int NCHUNK = (int)(NTOT / 64);           // 73728 wmma chunks
constexpr int NPART = 480;                         // reduction blocks
constexpr int TCOLS = 32;                          // column-tile width
#define BIGF 1.0e6f
#define BIG2F 1.0e12f

typedef __attribute__((ext_vector_type(2))) float v2f;
typedef __attribute__((ext_vector_type(8))) float v8f;
typedef __attribute__((ext_vector_type(4))) int v4i;

#define AS1 __attribute__((address_space(1)))
#define AS3 __attribute__((address_space(3)))

#if defined(__has_builtin)
#if __has_builtin(__builtin_amdgcn_global_load_async_to_lds_b128)
#define HAS_ASYNC 1
#endif
#if __has_builtin(__builtin_amdgcn_wmma_f32_16x16x4_f32)
#define HAS_WMMA4 1
#endif
#endif

__device__ __forceinline__ void async_copy16(float* dst_lds, const float* src) {
#ifdef HAS_ASYNC
  // builtin signature: (v4i addrspace(1)*, v4i addrspace(3)*, imm offset, imm cpol)
  __builtin_amdgcn_global_load_async_to_lds_b128(
      (AS1 v4i*)src, (AS3 v4i*)dst_lds, 0, 0);
#else
  *(float4*)dst_lds = *(const float4*)src;
#endif
}

__device__ __forceinline__ void wait_async_all() {
#ifdef HAS_ASYNC
  asm volatile("s_wait_asynccnt 0x0" ::: "memory");
#endif
}

// ---------------------------------------------------------------------------
// Kernel 1: exact 1D EDT along W (forward+backward scan), one wave per row.
// mode==1: mask = target > 0.5 ; mode==0: mask = pred > 0 (== sigmoid>0.5).
// Writes squared clamped distance d^2 into g2 (row-major, in place per row).
// Uses wave32 ballot so each chunk of 32 pixels is one coalesced step.
// ---------------------------------------------------------------------------
__global__ void __launch_bounds__(256) k_rowpass(const float* __restrict__ in,
                                                 float* __restrict__ g2, int mode) {
  const int lane = threadIdx.x & 31;
  const int row = blockIdx.x * 8 + (threadIdx.x >> 5);
  if (row >= ROWS) return;
  const float* rin = in + (size_t)row * Ww;
  float* rout = g2 + (size_t)row * Ww;

  // forward scan: f_j = (mask_j ? f_{j-1}+1 : 0), init BIG
  float carry = BIGF;
  for (int ch = 0; ch < Ww / 32; ++ch) {
    const float x = rin[ch * 32 + lane];
    const bool m = mode ? (x > 0.5f) : (x > 0.0f);
    const unsigned bal = __builtin_amdgcn_ballot_w32(m);
    const unsigned nz = ~bal;  // positions of False pixels in this chunk
    const unsigned le = (lane == 31) ? 0xFFFFFFFFu : ((1u << (lane + 1)) - 1u);
    const unsigned nzle = nz & le;
    float d;
    if (nzle)
      d = (float)(lane - (31 - __builtin_clz(nzle)));
    else
      d = carry + (float)(lane + 1);
    rout[ch * 32 + lane] = d;
    carry = __shfl(d, 31, 32);
  }

  // backward scan + combine + square
  carry = BIGF;
  for (int ch = Ww / 32 - 1; ch >= 0; --ch) {
    const float x = rin[ch * 32 + lane];
    const bool m = mode ? (x > 0.5f) : (x > 0.0f);
    const unsigned bal = __builtin_amdgcn_ballot_w32(m);
    const unsigned nz = ~bal;
    const unsigned ge = 0xFFFFFFFFu << lane;
    const unsigned nzge = nz & ge;
    float b;
    if (nzge)
      b = (float)(__builtin_ctz(nzge) - lane);
    else
      b = carry + (float)(32 - lane);
    const float f = rout[ch * 32 + lane];
    const float d = fminf(fminf(f, b), BIGF);
    rout[ch * 32 + lane] = d * d;
    carry = __shfl(b, 0, 32);
  }
}

// ---------------------------------------------------------------------------
// Kernel 2: column min-plus pass, in place.
//   D[i,j] = min(BIG^2, min_k g2[k,j] + (i-k)^2)
// One block per (slice, 32-column tile): async-stage the 384x32 f32 tile
// (48 KB) into LDS, then each thread owns one column x 48 output rows,
// keeping 6 running minima live per k-sweep. LDS reads are conflict-free
// (lane index == column index across 64 banks).
// ---------------------------------------------------------------------------
__global__ void __launch_bounds__(256) k_colpass(float* __restrict__ g2) {
  __shared__ __align__(16) float tile[Hh * TCOLS];  // 48 KB

  const int slice = blockIdx.x / (Ww / TCOLS);
  const int j0 = (blockIdx.x % (Ww / TCOLS)) * TCOLS;
  float* base = g2 + (size_t)slice * Hh * Ww;

  // stage tile: 3072 float4s, 12 per thread, coalesced 4 x 128B rows per wave
  constexpr int NF4 = Hh * TCOLS / 4;
  for (int e4 = threadIdx.x; e4 < NF4; e4 += 256) {
    const int c4 = e4 & (TCOLS / 4 - 1);  // float4 within row
    const int k = e4 >> 3;                // row
    async_copy16(&tile[k * TCOLS + c4 * 4], base + (size_t)k * Ww + j0 + c4 * 4);
  }
  wait_async_all();
  __syncthreads();

  const int col = threadIdx.x & 31;
  const int rg = threadIdx.x >> 5;  // 0..7 -> rows [rg*48, rg*48+48)

  for (int r0 = 0; r0 < 48; r0 += 6) {
    const int i0 = rg * 48 + r0;
    float d0 = BIG2F, d1 = BIG2F, d2 = BIG2F, d3 = BIG2F, d4 = BIG2F, d5 = BIG2F;
    float tb = (float)i0;  // (i0 - k)
#pragma unroll 4
    for (int k = 0; k < Hh; ++k) {
      const float g = tile[k * TCOLS + col];
      const float t0 = tb;
      const float t1 = tb + 1.0f;
      const float t2 = tb + 2.0f;
      const float t3 = tb + 3.0f;
      const float t4 = tb + 4.0f;
      const float t5 = tb + 5.0f;
      d0 = fminf(d0, fmaf(t0, t0, g));
      d1 = fminf(d1, fmaf(t1, t1, g));
      d2 = fminf(d2, fmaf(t2, t2, g));
      d3 = fminf(d3, fmaf(t3, t3, g));
      d4 = fminf(d4, fmaf(t4, t4, g));
      d5 = fminf(d5, fmaf(t5, t5, g));
      tb -= 1.0f;
    }
    float* o = base + (size_t)i0 * Ww + j0 + col;
    o[0 * Ww] = d0;
    o[1 * Ww] = d1;
    o[2 * Ww] = d2;
    o[3 * Ww] = d3;
    o[4 * Ww] = d4;
    o[5 * Ww] = d5;
  }
}

// ---------------------------------------------------------------------------
// Kernel 3: partial reduction of sum( (sigmoid(p)-t)^2 * D ) using the matrix
// unit: per wave, A = 64 loss terms (v2f per lane), B = ones, C accumulates
// (v_wmma_f32_16x16x4_f32). Sum of all 256 accumulator entries = 16 * true
// sum, so each partial is scaled by 1/16. Deterministic (no float atomics).
// ---------------------------------------------------------------------------
__global__ void __launch_bounds__(256) k_reduce(const float* __restrict__ pred,
                                                const float* __restrict__ targ,
                                                const float* __restrict__ D,
                                                float* __restrict__ partials,
                                                int nchunks) {
  const int lane = threadIdx.x & 31;
  const int wib = threadIdx.x >> 5;  // wave in block (0..7)
  const int wave = blockIdx.x * 8 + wib;
  const int totalWaves = gridDim.x * 8;

  float s;
#ifdef HAS_WMMA4
  v8f acc = 0.0f;
  const v2f ones = {1.0f, 1.0f};
  for (int cb = wave; cb < nchunks; cb += totalWaves) {
    const size_t idx = (size_t)cb * 64 + (size_t)lane * 2;
    __builtin_prefetch(&D[idx + 8192], 0, 0);
    const float p0 = 1.0f / (1.0f + __expf(-pred[idx]));
    const float p1 = 1.0f / (1.0f + __expf(-pred[idx + 1]));
    const float q0 = p0 - targ[idx];
    const float q1 = p1 - targ[idx + 1];
    v2f a;
    a.x = q0 * q0 * D[idx];
    a.y = q1 * q1 * D[idx + 1];
    acc = __builtin_amdgcn_wmma_f32_16x16x4_f32(false, a, false, ones,
                                                (short)0, acc, false, false);
  }
  s = 0.0f;
#pragma unroll
  for (int i = 0; i < 8; ++i) s += acc[i];
  s *= (1.0f / 16.0f);
#else
  s = 0.0f;
  for (int cb = wave; cb < nchunks; cb += totalWaves) {
    const size_t idx = (size_t)cb * 64 + (size_t)lane * 2;
    const float p0 = 1.0f / (1.0f + __expf(-pred[idx]));
    const float p1 = 1.0f / (1.0f + __expf(-pred[idx + 1]));
    const float q0 = p0 - targ[idx];
    const float q1 = p1 - targ[idx + 1];
    s += q0 * q0 * D[idx] + q1 * q1 * D[idx + 1];
  }
#endif
  // wave reduce
  for (int off = 16; off; off >>= 1) s += __shfl_down(s, off, 32);
  __shared__ float wsum[8];
  if (lane == 0) wsum[wib] = s;
  __syncthreads();
  if (threadIdx.x == 0) {
    float t = 0.0f;
#pragma unroll
    for (int w = 0; w < 8; ++w) t += wsum[w];
    partials[blockIdx.x] = t;
  }
}

// ---------------------------------------------------------------------------
// Kernel 4: deterministic finalize: out = (sum(partA) + sum(partB)) / N
// ---------------------------------------------------------------------------
__global__ void __launch_bounds__(256) k_final(const float* __restrict__ pa,
                                               const float* __restrict__ pb,
                                               float* __restrict__ out,
                                               int npart, float invN) {
  __shared__ float sm[256];
  float t = 0.0f;
  for (int i = threadIdx.x; i < npart; i += 256) t += pa[i] + pb[i];
  sm[threadIdx.x] = t;
  __syncthreads();
  for (int st = 128; st; st >>= 1) {
    if (threadIdx.x < st) sm[threadIdx.x] += sm[threadIdx.x + st];
    __syncthreads();
  }
  if (threadIdx.x == 0) out[0] = sm[0] * invN;
}

// ---------------------------------------------------------------------------
extern "C" void kernel_launch(void* const* d_in, const int* in_sizes, int n_in,
                              void* d_out, int out_size, void* d_ws, size_t ws_size,
                              hipStream_t stream) {
  const float* pred = (const float*)d_in[0];
  const float* targ = (const float*)d_in[1];
  float* out = (float*)d_out;

  float* g2 = (float*)d_ws;         // NTOT floats: squared EDT scratch (reused)
  float* partA = g2 + NTOT;         // NPART floats (gt term)
  float* partB = partA + NPART;     // NPART floats (seg term)

  const dim3 blk(256);
  const dim3 gRow(ROWS / 8);                 // 1536 blocks, 8 waves each
  const dim3 gCol(SL * (Ww / TCOLS));        // 384 blocks
  const dim3 gRed(NPART);

  // ---- ground-truth distance term: sum delta * D_gt -----------------------
  k_rowpass<<<gRow, blk, 0, stream>>>(targ, g2, /*mode=gt*/ 1);
  k_colpass<<<gCol, blk, 0, stream>>>(g2);
  k_reduce<<<gRed, blk, 0, stream>>>(pred, targ, g2, partA, NCHUNK);

  // ---- segmentation distance term: sum delta * D_seg ----------------------
  k_rowpass<<<gRow, blk, 0, stream>>>(pred, g2, /*mode=seg*/ 0);
  k_colpass<<<gCol, blk, 0, stream>>>(g2);
  k_reduce<<<gRed, blk, 0, stream>>>(pred, targ, g2, partB, NCHUNK);

  // ---- finalize mean ------------------------------------------------------
  k_final<<<1, blk, 0, stream>>>(partA, partB, out, NPART, 1.0f / (float)NTOT);
}